// TemporalGNN_46093589021344
// MI455X (gfx1250) — compile-verified
//
#include <hip/hip_runtime.h>
#include <cstdint>
#include <cstddef>

#define N_NODES 20000
#define N_EDGES 320000
#define F_IN    128
#define HDIM    32
#define K_CH    32
#define D_EMB   128
#define NEG_SLOPE 0.2f

typedef __attribute__((ext_vector_type(2))) float v2f;
typedef __attribute__((ext_vector_type(8))) float v8f;

__device__ __forceinline__ float leaky(float x) { return x > 0.f ? x : NEG_SLOPE * x; }

// ---------------- utility fills ----------------
__global__ void k_fill_f32(float* p, float v, int n) {
    int i = blockIdx.x * blockDim.x + threadIdx.x;
    if (i < n) p[i] = v;
}
__global__ void k_fill_i32(int* p, int v, int n) {
    int i = blockIdx.x * blockDim.x + threadIdx.x;
    if (i < n) p[i] = v;
}
__global__ void k_bias_rows(float* p, const float* bias, int n) {   // p[i] = bias[i % 128]
    int i = blockIdx.x * blockDim.x + threadIdx.x;
    if (i < n) p[i] = bias[i & (D_EMB - 1)];
}

// ---------------- graph preprocessing ----------------
__global__ void k_deg(const int* __restrict__ src, float* deg, int e) {
    int i = blockIdx.x * blockDim.x + threadIdx.x;
    if (i < e) atomicAdd(&deg[src[i]], 1.0f);
}
__global__ void k_dis(float* deg, int n) {   // in place: deg -> 1/sqrt(deg), 0 if deg==0
    int i = blockIdx.x * blockDim.x + threadIdx.x;
    if (i < n) { float d = deg[i]; deg[i] = d > 0.f ? 1.0f / sqrtf(d) : 0.f; }
}
__global__ void k_lw(const int* __restrict__ src, const int* __restrict__ dst,
                     const float* __restrict__ dis, float* lw, int e) {
    int i = blockIdx.x * blockDim.x + threadIdx.x;
    if (i < e) lw[i] = -dis[src[i]] * dis[dst[i]];
}
__global__ void k_count(const int* __restrict__ dst, int* cnt, int e) {
    int i = blockIdx.x * blockDim.x + threadIdx.x;
    if (i < e) atomicAdd(&cnt[dst[i]], 1);
}
// single-workgroup exclusive scan over N (20 chunks of 1024)
__global__ void k_scan(const int* __restrict__ cnt, int* row_ptr, int* cursor, int n) {
    __shared__ int sh[1024];
    __shared__ int carry_sh;
    int t = threadIdx.x;
    if (t == 0) { carry_sh = 0; row_ptr[0] = 0; }
    __syncthreads();
    for (int base = 0; base < n; base += 1024) {
        int i = base + t;
        int v = (i < n) ? cnt[i] : 0;
        sh[t] = v;
        __syncthreads();
        for (int off = 1; off < 1024; off <<= 1) {
            int add = (t >= off) ? sh[t - off] : 0;
            __syncthreads();
            sh[t] += add;
            __syncthreads();
        }
        int incl = sh[t];
        int carry = carry_sh;
        if (i < n) { row_ptr[i + 1] = carry + incl; cursor[i] = carry + incl - v; }
        __syncthreads();
        if (t == 1023) carry_sh = carry + incl;
        __syncthreads();
    }
}
__global__ void k_fill_csr(const int* __restrict__ src, const int* __restrict__ dst,
                           const float* __restrict__ lw, int* cursor,
                           int* col_src, float* col_w, int e) {
    int i = blockIdx.x * blockDim.x + threadIdx.x;
    if (i < e) {
        int p = atomicAdd(&cursor[dst[i]], 1);
        col_src[p] = src[i];
        col_w[p]   = lw[i];
    }
}
// concat W_xz[k] | W_xh[k] into [K][128][64]
__global__ void k_pack_wc(const float* __restrict__ Wxz, const float* __restrict__ Wxh,
                          float* Wc, int total) {
    int i = blockIdx.x * blockDim.x + threadIdx.x;
    if (i >= total) return;
    int k = i / (F_IN * 64);
    int r = i - k * (F_IN * 64);
    int f = r >> 6;
    int c = r & 63;
    Wc[i] = (c < HDIM) ? Wxz[(k * F_IN + f) * HDIM + c]
                       : Wxh[(k * F_IN + f) * HDIM + (c - HDIM)];
}

// ---------------- Chebyshev propagation (CSR gather, wave per node) ----------------
// Tn[node] = sA * sum_j w_j * Tcur[src_j] + sB * Tprev[node]
__global__ void k_prop(const int* __restrict__ rp, const int* __restrict__ cs,
                       const float* __restrict__ cw,
                       const float* __restrict__ Tcur, const float* __restrict__ Tprev,
                       float* __restrict__ Tn, float sA, float sB, int n) {
    int gw   = (blockIdx.x * blockDim.x + threadIdx.x) >> 5;
    int lane = threadIdx.x & 31;
    if (gw >= n) return;
    float4 acc; acc.x = acc.y = acc.z = acc.w = 0.f;
    int beg = rp[gw], end = rp[gw + 1];
    const float4* base = (const float4*)Tcur;
    for (int j = beg; j < end; ++j) {
        float  w = cw[j];
        float4 v = base[(size_t)cs[j] * 32 + lane];
        acc.x += w * v.x; acc.y += w * v.y; acc.z += w * v.z; acc.w += w * v.w;
    }
    float4 o;
    if (sB != 0.f) {
        float4 pv = ((const float4*)Tprev)[(size_t)gw * 32 + lane];
        o.x = sA * acc.x + sB * pv.x; o.y = sA * acc.y + sB * pv.y;
        o.z = sA * acc.z + sB * pv.z; o.w = sA * acc.w + sB * pv.w;
    } else {
        o.x = sA * acc.x; o.y = sA * acc.y; o.z = sA * acc.z; o.w = sA * acc.w;
    }
    ((float4*)Tn)[(size_t)gw * 32 + lane] = o;
}

// ---------------- WMMA f32 GEMM: C[M,NCOLS] += A[M,KDIM] * B[KDIM,NCOLS] ----------------
// wave32, V_WMMA_F32_16X16X4_F32. One wave per 16-row tile; B staged in LDS.
template <int KDIM, int NCOLS>
__global__ void k_gemm(const float* __restrict__ A, const float* __restrict__ B,
                       float* __restrict__ C, int mtiles) {
    __shared__ float bsh[KDIM * NCOLS];
    for (int i = threadIdx.x; i < KDIM * NCOLS; i += blockDim.x) bsh[i] = B[i];
    __syncthreads();

    int wid  = (blockIdx.x * blockDim.x + threadIdx.x) >> 5;
    int lane = threadIdx.x & 31;
    if (wid >= mtiles) return;

    int row0  = wid << 4;
    int rlo   = lane & 15;
    int khalf = (lane >> 4) << 1;              // 0 (lanes 0-15) or 2 (lanes 16-31)

    // A frags for all K steps: a.x = A[row][k0+khalf], a.y = A[row][k0+khalf+1]
    const float* arow = A + (size_t)(row0 + rlo) * KDIM + khalf;
    v2f af[KDIM / 4];
#pragma unroll
    for (int kk = 0; kk < KDIM / 4; ++kk) af[kk] = *(const v2f*)(arow + kk * 4);

    int rowc = row0 + ((lane >> 4) << 3);      // C: vgpr r -> row rowc+r, col = nt+rlo
#pragma unroll
    for (int nt = 0; nt < NCOLS; nt += 16) {
        int col = nt + rlo;
        v8f c;
#pragma unroll
        for (int r = 0; r < 8; ++r) c[r] = C[(size_t)(rowc + r) * NCOLS + col];
#pragma unroll
        for (int kk = 0; kk < KDIM / 4; ++kk) {
            int ks = kk * 4 + khalf;
            v2f b;
            b.x = bsh[ks * NCOLS + col];
            b.y = bsh[(ks + 1) * NCOLS + col];
            c = __builtin_amdgcn_wmma_f32_16x16x4_f32(false, af[kk], false, b,
                                                      (short)0, c, false, false);
        }
#pragma unroll
        for (int r = 0; r < 8; ++r) C[(size_t)(rowc + r) * NCOLS + col] = c[r];
    }
}

// ---------------- GRU gate fusion: h = (1 - sigmoid(z)) * tanh(ht) ----------------
__global__ void k_gates(const float* __restrict__ acc,
                        const float* __restrict__ bxz, const float* __restrict__ bhz,
                        const float* __restrict__ bxh, const float* __restrict__ bhh,
                        float* __restrict__ h, int total) {
    int i = blockIdx.x * blockDim.x + threadIdx.x;
    if (i >= total) return;
    int n = i >> 5, f = i & 31;
    float zin = acc[(size_t)n * 64 + f] + bxz[f] + bhz[f];
    float hin = acc[(size_t)n * 64 + 32 + f] + bxh[f] + bhh[f];
    float z = 1.f / (1.f + expf(-zin));
    h[i] = (1.f - z) * tanhf(hin);
}

// ---------------- GAT attention coefficients (wave per node) ----------------
__global__ void k_attvec(const float* __restrict__ xp, const float* __restrict__ avs,
                         const float* __restrict__ avd, float* asv, float* adv, int n) {
    int gw   = (blockIdx.x * blockDim.x + threadIdx.x) >> 5;
    int lane = threadIdx.x & 31;
    if (gw >= n) return;
    float xv = xp[(size_t)gw * 32 + lane];
    float s  = xv * avs[lane];
    float d  = xv * avd[lane];
    for (int off = 16; off; off >>= 1) { s += __shfl_xor(s, off); d += __shfl_xor(d, off); }
    if (lane == 0) { asv[gw] = s; adv[gw] = d; }
}

// ---------------- GAT softmax-aggregate over incoming edges + self loop ----------------
__global__ void k_gat(const int* __restrict__ rp, const int* __restrict__ cs,
                      const float* __restrict__ asv, const float* __restrict__ adv,
                      const float* __restrict__ xp, const float* __restrict__ bgat,
                      float* __restrict__ h2, int n) {
    int gw   = (blockIdx.x * blockDim.x + threadIdx.x) >> 5;
    int lane = threadIdx.x & 31;
    if (gw >= n) return;
    float adn    = adv[gw];
    float self_e = leaky(asv[gw] + adn);
    int beg = rp[gw], end = rp[gw + 1];
    // segment max (lane-parallel over edges, includes self loop)
    float mx = self_e;
    for (int j = beg + lane; j < end; j += 32) mx = fmaxf(mx, leaky(asv[cs[j]] + adn));
    for (int off = 16; off; off >>= 1) mx = fmaxf(mx, __shfl_xor(mx, off));
    // weighted feature aggregation, lane = feature
    float facc = 0.f, dsum = 0.f;
    for (int j = beg; j < end; ++j) {
        int   s = cs[j];
        float w = expf(leaky(asv[s] + adn) - mx);
        dsum += w;
        facc += w * xp[(size_t)s * 32 + lane];
    }
    float ws = expf(self_e - mx);
    dsum += ws;
    facc += ws * xp[(size_t)gw * 32 + lane];
    float hv = facc / dsum + bgat[lane];
    h2[(size_t)gw * 32 + lane] = fmaxf(hv, 0.f);
}

// ---------------- edge classifier: logit[e] = [emb[src]|emb[dst]] . W_cls + b ----------------
__global__ void k_cls(const int* __restrict__ src, const int* __restrict__ dst,
                      const float* __restrict__ emb, const float* __restrict__ Wc,
                      const float* __restrict__ bc, float* __restrict__ out, int e) {
    int i = blockIdx.x * blockDim.x + threadIdx.x;
    if (i >= e) return;
    const float4* ps = (const float4*)(emb + (size_t)src[i] * D_EMB);
    const float4* pd = (const float4*)(emb + (size_t)dst[i] * D_EMB);
    const float4* w0 = (const float4*)(Wc);
    const float4* w1 = (const float4*)(Wc + D_EMB);
    float acc = bc[0];
#pragma unroll
    for (int k = 0; k < D_EMB / 4; ++k) {
        float4 a = ps[k], w = w0[k];
        acc += a.x * w.x + a.y * w.y + a.z * w.z + a.w * w.w;
    }
#pragma unroll
    for (int k = 0; k < D_EMB / 4; ++k) {
        float4 a = pd[k], w = w1[k];
        acc += a.x * w.x + a.y * w.y + a.z * w.z + a.w * w.w;
    }
    out[i] = acc;
}

// ---------------- host launcher ----------------
extern "C" void kernel_launch(void* const* d_in, const int* in_sizes, int n_in,
                              void* d_out, int out_size, void* d_ws, size_t ws_size,
                              hipStream_t stream) {
    (void)in_sizes; (void)n_in; (void)out_size; (void)ws_size;
    const float* x     = (const float*)d_in[0];
    const int*   edge  = (const int*)d_in[1];
    const int*   src   = edge;
    const int*   dst   = edge + N_EDGES;
    const float* W_xz  = (const float*)d_in[2];
    const float* W_xh  = (const float*)d_in[6];
    const float* b_xz  = (const float*)d_in[8];
    const float* b_hz  = (const float*)d_in[9];
    const float* b_xh  = (const float*)d_in[12];
    const float* b_hh  = (const float*)d_in[13];
    const float* W_gat = (const float*)d_in[14];
    const float* a_src = (const float*)d_in[15];
    const float* a_dst = (const float*)d_in[16];
    const float* b_gat = (const float*)d_in[17];
    const float* W_emb = (const float*)d_in[18];
    const float* b_emb = (const float*)d_in[19];
    const float* W_cls = (const float*)d_in[20];
    const float* b_cls = (const float*)d_in[21];
    float* out = (float*)d_out;

    // workspace carve (256B aligned)
    uint8_t* wp = (uint8_t*)d_ws;
    auto carve = [&](size_t bytes) -> void* {
        void* p = (void*)wp;
        wp += (bytes + 255) & ~(size_t)255;
        return p;
    };
    float* deg    = (float*)carve((size_t)N_NODES * 4);
    float* lw     = (float*)carve((size_t)N_EDGES * 4);
    int*   indeg  = (int*)carve((size_t)N_NODES * 4);
    int*   rp     = (int*)carve((size_t)(N_NODES + 1) * 4);
    int*   cursor = (int*)carve((size_t)N_NODES * 4);
    int*   csrc   = (int*)carve((size_t)N_EDGES * 4);
    float* cw     = (float*)carve((size_t)N_EDGES * 4);
    float* T0     = (float*)carve((size_t)N_NODES * F_IN * 4);
    float* T1     = (float*)carve((size_t)N_NODES * F_IN * 4);
    float* T2     = (float*)carve((size_t)N_NODES * F_IN * 4);
    float* accZH  = (float*)carve((size_t)N_NODES * 64 * 4);
    float* Wc     = (float*)carve((size_t)K_CH * F_IN * 64 * 4);
    float* h      = (float*)carve((size_t)N_NODES * HDIM * 4);
    float* xp     = (float*)carve((size_t)N_NODES * HDIM * 4);
    float* asv    = (float*)carve((size_t)N_NODES * 4);
    float* adv    = (float*)carve((size_t)N_NODES * 4);
    float* h2     = (float*)carve((size_t)N_NODES * HDIM * 4);
    float* emb    = (float*)carve((size_t)N_NODES * D_EMB * 4);

    const int TB = 256;
    const int gN   = (N_NODES + TB - 1) / TB;
    const int gE   = (N_EDGES + TB - 1) / TB;
    const int gNW  = (N_NODES * 32 + TB - 1) / TB;      // wave-per-node kernels
    const int MT   = N_NODES / 16;                      // 1250 row tiles
    const int gGEM = (MT * 32 + TB - 1) / TB;           // one wave per tile

    // --- graph normalization + CSR build ---
    k_fill_f32<<<gN, TB, 0, stream>>>(deg, 0.f, N_NODES);
    k_deg<<<gE, TB, 0, stream>>>(src, deg, N_EDGES);
    k_dis<<<gN, TB, 0, stream>>>(deg, N_NODES);
    k_lw<<<gE, TB, 0, stream>>>(src, dst, deg, lw, N_EDGES);
    k_fill_i32<<<gN, TB, 0, stream>>>(indeg, 0, N_NODES);
    k_count<<<gE, TB, 0, stream>>>(dst, indeg, N_EDGES);
    k_scan<<<1, 1024, 0, stream>>>(indeg, rp, cursor, N_NODES);
    k_fill_csr<<<gE, TB, 0, stream>>>(src, dst, lw, cursor, csrc, cw, N_EDGES);

    // --- pack fused Cheb weights, zero accumulator ---
    k_pack_wc<<<(K_CH * F_IN * 64 + TB - 1) / TB, TB, 0, stream>>>(W_xz, W_xh, Wc,
                                                                   K_CH * F_IN * 64);
    k_fill_f32<<<(N_NODES * 64 + TB - 1) / TB, TB, 0, stream>>>(accZH, 0.f, N_NODES * 64);

    // --- shared Chebyshev recurrence: T0 = x, T1 = L_hat x, Tk = 2 L_hat T(k-1) - T(k-2)
    hipMemcpyAsync(T0, x, (size_t)N_NODES * F_IN * 4, hipMemcpyDeviceToDevice, stream);
    k_prop<<<gNW, TB, 0, stream>>>(rp, csrc, cw, T0, T0, T1, 1.f, 0.f, N_NODES);
    k_gemm<F_IN, 64><<<gGEM, TB, 0, stream>>>(T0, Wc, accZH, MT);
    k_gemm<F_IN, 64><<<gGEM, TB, 0, stream>>>(T1, Wc + (size_t)F_IN * 64, accZH, MT);
    {
        float* T[3] = {T0, T1, T2};
        int pi = 0, ci = 1;
        for (int k = 2; k < K_CH; ++k) {
            int ni = 3 - pi - ci;
            k_prop<<<gNW, TB, 0, stream>>>(rp, csrc, cw, T[ci], T[pi], T[ni], 2.f, -1.f,
                                           N_NODES);
            k_gemm<F_IN, 64><<<gGEM, TB, 0, stream>>>(T[ni], Wc + (size_t)k * F_IN * 64,
                                                      accZH, MT);
            pi = ci; ci = ni;
        }
    }

    // --- GRU gates (H0 = 0 => h = (1-Z)*Ht) ---
    k_gates<<<gNW, TB, 0, stream>>>(accZH, b_xz, b_hz, b_xh, b_hh, h, N_NODES * HDIM);

    // --- GAT ---
    k_fill_f32<<<gNW, TB, 0, stream>>>(xp, 0.f, N_NODES * HDIM);
    k_gemm<HDIM, 32><<<gGEM, TB, 0, stream>>>(h, W_gat, xp, MT);
    k_attvec<<<gNW, TB, 0, stream>>>(xp, a_src, a_dst, asv, adv, N_NODES);
    k_gat<<<gNW, TB, 0, stream>>>(rp, csrc, asv, adv, xp, b_gat, h2, N_NODES);

    // --- embedding + edge classifier ---
    k_bias_rows<<<(N_NODES * D_EMB + TB - 1) / TB, TB, 0, stream>>>(emb, b_emb,
                                                                    N_NODES * D_EMB);
    k_gemm<HDIM, D_EMB><<<gGEM, TB, 0, stream>>>(h2, W_emb, emb, MT);
    k_cls<<<gE, TB, 0, stream>>>(src, dst, emb, W_cls, b_cls, out, N_EDGES);
}